// KANLayer_64802466562625
// MI455X (gfx1250) — compile-verified
//
#include <hip/hip_runtime.h>

// KAN layer for MI455X (gfx1250, wave32).
// Memory-bound problem: interp tensor (131 MB fp32) is L2-resident (192 MB L2),
// gather streams ~1 GB from L2 (now with b128 loads); interp construction is a
// blocked WMMA scan using V_WMMA_F32_16X16X4_F32.

#define IN_F   128
#define OUT_F  256
#define NCP    1000
#define BATCH  4096
#define EPSV   1e-6f

typedef __attribute__((ext_vector_type(2))) float v2f;
typedef __attribute__((ext_vector_type(8))) float v8f;

// ---------------------------------------------------------------- K0: min/max
__global__ void __launch_bounds__(256) kan_minmax(const float* __restrict__ x,
                                                  float* __restrict__ mins,
                                                  float* __restrict__ maxs) {
  __shared__ float smin[256];
  __shared__ float smax[256];
  const int i = blockIdx.x;                       // feature
  float mn = 3.4e38f, mx = -3.4e38f;
  for (int b = threadIdx.x; b < BATCH; b += 256) {
    float v = x[(size_t)b * IN_F + i];
    mn = fminf(mn, v);
    mx = fmaxf(mx, v);
  }
  smin[threadIdx.x] = mn;
  smax[threadIdx.x] = mx;
  __syncthreads();
  for (int s = 128; s > 0; s >>= 1) {
    if (threadIdx.x < s) {
      smin[threadIdx.x] = fminf(smin[threadIdx.x], smin[threadIdx.x + s]);
      smax[threadIdx.x] = fmaxf(smax[threadIdx.x], smax[threadIdx.x + s]);
    }
    __syncthreads();
  }
  if (threadIdx.x == 0) { mins[i] = smin[0]; maxs[i] = smax[0]; }
}

// ------------------------------------------------- K1: l-weight totals per (i,o)
// ltot  = sum_c l[i,c,o]          lbtot = sum_c c * l[i,c,o]
__global__ void __launch_bounds__(256) kan_ltot(const float* __restrict__ lw,
                                                float* __restrict__ ltot,
                                                float* __restrict__ lbtot) {
  const int idx = blockIdx.x * 256 + threadIdx.x;   // i*OUT_F + o
  const int i = idx / OUT_F;
  const int o = idx % OUT_F;
  const float* p = lw + (size_t)i * NCP * OUT_F + o;
  float s = 0.f, sb = 0.f;
  for (int c = 0; c < NCP; ++c) {
    float v = p[(size_t)c * OUT_F];
    s += v;
    sb = fmaf((float)c, v, sb);
  }
  ltot[idx] = s;
  lbtot[idx] = sb;
}

// ---------------------------------------------------------- K2: bucket + frac
__global__ void __launch_bounds__(256) kan_bucket(const float* __restrict__ x,
                                                  const float* __restrict__ mins,
                                                  const float* __restrict__ maxs,
                                                  int* __restrict__ lo,
                                                  float* __restrict__ frac) {
  const int idx = blockIdx.x * 256 + threadIdx.x;   // b*IN_F + i
  const int i = idx & (IN_F - 1);
  float xs = (x[idx] - mins[i]) / (maxs[i] - mins[i] + EPSV) * (float)(NCP - 1);
  float lf = fminf(fmaxf(floorf(xs), 0.f), (float)(NCP - 2));
  lo[idx] = (int)lf;
  frac[idx] = xs - lf;
}

// ----------------------------------------------- K3: interp build (WMMA scan)
// interp[i,c,o] = sum_{c'<=c}(c-c') r[i,c',o] + sum_{c'>c}(c'-c) l[i,c',o]
// Blocked scan over 16-wide chunks of c. Per chunk (16 rows x 16 cols tile):
//   D = M @ R + M' @ L + (Pr + Ql0) + m*(Ar - AlF)      (rank-1 carry)
// with fixed M[m,k]=max(m-k,0), M'[m,k]=max(k-m,0); all matmuls via
// V_WMMA_F32_16X16X4_F32 (4 k-chunks each). Column sums needed for the carry
// update are produced by a "stats" WMMA (A row0 = ones, A row8 = k) and
// exchanged across half-waves with shfl_xor(16).
__global__ void __launch_bounds__(256) kan_interp_wmma(
    const float* __restrict__ rw, const float* __restrict__ lw,
    const float* __restrict__ ltot, const float* __restrict__ lbtot,
    float* __restrict__ interp) {
  const int lane = threadIdx.x & 31;
  const int wid  = threadIdx.x >> 5;
  const int wg   = blockIdx.x * 8 + wid;     // 0..2047 waves
  const int i    = wg >> 4;                  // feature 0..127
  const int ot   = wg & 15;                  // output 16-tile 0..15
  const int m    = lane & 15;                // A row, B/C column
  const int half = lane >> 4;                // 0: K=0,1  1: K=2,3
  const int ocol = ot * 16 + m;              // global output column

  // Constant A operands (16x4 f32: lane m holds A[m, k0], A[m, k1])
  v2f MA[4], MP[4], SA[4];
#pragma unroll
  for (int kk = 0; kk < 4; ++kk) {
    const int k0 = kk * 4 + half * 2, k1 = k0 + 1;
    MA[kk].x = (float)((m > k0) ? (m - k0) : 0);
    MA[kk].y = (float)((m > k1) ? (m - k1) : 0);
    MP[kk].x = (float)((k0 > m) ? (k0 - m) : 0);
    MP[kk].y = (float)((k1 > m) ? (k1 - m) : 0);
    SA[kk].x = (m == 0) ? 1.f : ((m == 8) ? (float)k0 : 0.f);
    SA[kk].y = (m == 0) ? 1.f : ((m == 8) ? (float)k1 : 0.f);
  }

  const size_t fb = (size_t)i * NCP * OUT_F + ocol;
  const float* Rp = rw + fb;
  const float* Lp = lw + fb;
  float* Ip = interp + fb;
  const float Lt  = ltot [i * OUT_F + ocol];
  const float Lbt = lbtot[i * OUT_F + ocol];

  float Ar = 0.f;      // sum_{c'<cb} r
  float Pr = 0.f;      // sum_{c'<cb} (cb - c') r
  float preL  = 0.f;   // sum of l over processed chunks
  float preLb = 0.f;   // sum of c'*l over processed chunks

  for (int ch = 0; ch < 63; ++ch) {
    const int cb = ch * 16;
    const float cbf = (float)cb;

    // B operands (4x16 f32: lane n holds B[k0,n] / B[k1,n])
    v2f RB[4], LB[4];
    if (cb + 16 <= NCP) {
#pragma unroll
      for (int kk = 0; kk < 4; ++kk) {
        const int k0 = cb + kk * 4 + half * 2;
        RB[kk].x = Rp[(size_t)k0 * OUT_F];
        RB[kk].y = Rp[(size_t)(k0 + 1) * OUT_F];
        LB[kk].x = Lp[(size_t)k0 * OUT_F];
        LB[kk].y = Lp[(size_t)(k0 + 1) * OUT_F];
      }
    } else {  // last partial chunk: zero-pad c >= NCP
#pragma unroll
      for (int kk = 0; kk < 4; ++kk) {
        const int k0 = cb + kk * 4 + half * 2, k1 = k0 + 1;
        RB[kk].x = (k0 < NCP) ? Rp[(size_t)k0 * OUT_F] : 0.f;
        RB[kk].y = (k1 < NCP) ? Rp[(size_t)k1 * OUT_F] : 0.f;
        LB[kk].x = (k0 < NCP) ? Lp[(size_t)k0 * OUT_F] : 0.f;
        LB[kk].y = (k1 < NCP) ? Lp[(size_t)k1 * OUT_F] : 0.f;
      }
    }

    // stats(L): row0 -> colsum(L), row8 -> sum_k k*L[k,o]
    v8f sl = {};
#pragma unroll
    for (int kk = 0; kk < 4; ++kk)
      sl = __builtin_amdgcn_wmma_f32_16x16x4_f32(false, SA[kk], false, LB[kk],
                                                 (short)0, sl, false, false);
    float ownL = sl[0];
    float othL = __shfl_xor(ownL, 16, 32);
    const float SL0 = half ? othL : ownL;     // sum_k L[k,o]
    const float SLw = half ? ownL : othL;     // sum_k k*L[k,o]

    const float AlF  = Lt  - preL  - SL0;                 // sum_{c'>=cb+16} l
    const float QlbF = Lbt - preLb - fmaf(cbf, SL0, SLw); // sum_{c'>=cb+16} c' l
    const float Ql0  = QlbF - cbf * AlF;                  // sum (c'-cb) l

    // accumulator seeded with the rank-1 carry: C[m,o] = base + m*slope
    const float base  = Pr + Ql0;
    const float slope = Ar - AlF;
    v8f acc;
#pragma unroll
    for (int r = 0; r < 8; ++r)
      acc[r] = fmaf((float)(r + half * 8), slope, base);

#pragma unroll
    for (int kk = 0; kk < 4; ++kk)
      acc = __builtin_amdgcn_wmma_f32_16x16x4_f32(false, MA[kk], false, RB[kk],
                                                  (short)0, acc, false, false);
#pragma unroll
    for (int kk = 0; kk < 4; ++kk)
      acc = __builtin_amdgcn_wmma_f32_16x16x4_f32(false, MP[kk], false, LB[kk],
                                                  (short)0, acc, false, false);

    // store tile (guard only matters in the last partial chunk)
#pragma unroll
    for (int r = 0; r < 8; ++r) {
      const int c = cb + r + half * 8;
      if (c < NCP) Ip[(size_t)c * OUT_F] = acc[r];
    }

    // stats(R) and carry updates for next chunk
    v8f sr = {};
#pragma unroll
    for (int kk = 0; kk < 4; ++kk)
      sr = __builtin_amdgcn_wmma_f32_16x16x4_f32(false, SA[kk], false, RB[kk],
                                                 (short)0, sr, false, false);
    float ownR = sr[0];
    float othR = __shfl_xor(ownR, 16, 32);
    const float SR0 = half ? othR : ownR;
    const float SRw = half ? ownR : othR;

    Ar += SR0;
    Pr  = Pr + 16.f * Ar - SRw;     // Pr + 16*Ar_old + sum_k (16-k) R[k]
    preL  += SL0;
    preLb += fmaf(cbf, SL0, SLw);
  }
}

// --------------------------------------------------- K4: gather + lerp + sum
// 4 batch rows per 256-thread block; 64 threads per row, 4 outputs per thread
// via b128 loads -> 512 B of L2 traffic per load instruction, fully coalesced.
__global__ void __launch_bounds__(256) kan_gather(const float* __restrict__ interp,
                                                  const int* __restrict__ lo,
                                                  const float* __restrict__ frac,
                                                  float* __restrict__ out) {
  __shared__ int   slo[4][IN_F];
  __shared__ float sfr[4][IN_F];
  const int ro = threadIdx.x >> 6;          // row within block: 0..3
  const int t  = threadIdx.x & 63;          // 0..63
  const int b  = blockIdx.x * 4 + ro;

  // stage lo/frac for the 4 rows (512 entries, 2 per thread)
  for (int k = threadIdx.x; k < 4 * IN_F; k += 256) {
    const int rr = k >> 7;                  // k / 128
    const int ii = k & (IN_F - 1);
    const int bb = blockIdx.x * 4 + rr;
    slo[rr][ii] = lo[bb * IN_F + ii];
    sfr[rr][ii] = frac[bb * IN_F + ii];
  }
  __syncthreads();

  const int o = t * 4;
  float4 acc = make_float4(0.f, 0.f, 0.f, 0.f);
#pragma unroll 4
  for (int i = 0; i < IN_F; ++i) {
    const float* p = interp + ((size_t)i * NCP + slo[ro][i]) * OUT_F + o;
    const float4 v0 = *(const float4*)p;            // row lo
    const float4 v1 = *(const float4*)(p + OUT_F);  // row lo+1
    const float f = sfr[ro][i];
    acc.x += fmaf(f, v1.x - v0.x, v0.x);
    acc.y += fmaf(f, v1.y - v0.y, v0.y);
    acc.z += fmaf(f, v1.z - v0.z, v0.z);
    acc.w += fmaf(f, v1.w - v0.w, v0.w);
  }
  *(float4*)(out + (size_t)b * OUT_F + o) = acc;
}

// ---------------------------------------------------------------- launcher
extern "C" void kernel_launch(void* const* d_in, const int* in_sizes, int n_in,
                              void* d_out, int out_size, void* d_ws, size_t ws_size,
                              hipStream_t stream) {
  const float* x  = (const float*)d_in[0];   // (4096, 128)
  const float* rw = (const float*)d_in[1];   // (128, 1000, 256)
  const float* lw = (const float*)d_in[2];   // (128, 1000, 256)
  float* out = (float*)d_out;                // (4096, 256)

  // workspace layout (fp32 unless noted): ~136 MB total
  float* ws     = (float*)d_ws;
  float* interp = ws;                                          // 128*1000*256
  float* mins   = interp + (size_t)IN_F * NCP * OUT_F;         // 128
  float* maxs   = mins + IN_F;                                 // 128
  float* ltot   = maxs + IN_F;                                 // 128*256
  float* lbtot  = ltot + IN_F * OUT_F;                         // 128*256
  float* frac   = lbtot + IN_F * OUT_F;                        // 4096*128
  int*   lo     = (int*)(frac + (size_t)BATCH * IN_F);         // 4096*128

  kan_minmax<<<IN_F, 256, 0, stream>>>(x, mins, maxs);
  kan_ltot<<<(IN_F * OUT_F) / 256, 256, 0, stream>>>(lw, ltot, lbtot);
  kan_bucket<<<(BATCH * IN_F) / 256, 256, 0, stream>>>(x, mins, maxs, lo, frac);
  kan_interp_wmma<<<256, 256, 0, stream>>>(rw, lw, ltot, lbtot, interp);
  kan_gather<<<BATCH / 4, 256, 0, stream>>>(interp, lo, frac, out);
}